// Beta_SpikeSlab_Layer_Withoutbias_45165876085451
// MI455X (gfx1250) — compile-verified
//
#include <hip/hip_runtime.h>
#include <hip/hip_bf16.h>
#include <cstddef>

// MI455X (gfx1250) — wave32, WMMA. This problem is a 262144x1024 f32 GEMV
// (bandwidth bound: 1 GiB of x @ 23.3 TB/s ~ 46us floor), so we keep f32 end
// to end and use V_WMMA_F32_16X16X4_F32 for the reduction (16 rows/wave,
// K-chunks of 4, f32 accumulate). Precision-reducing WMMA flavors save no
// HBM bytes here.

typedef __attribute__((ext_vector_type(2))) float v2f;
typedef __attribute__((ext_vector_type(4))) float v4f;
typedef __attribute__((ext_vector_type(8))) float v8f;

#define D_DIM 1024
#define EPS 1e-7f
#define EPS_HC 1e-4f

__device__ __forceinline__ float softplusf(float x) {
    // numerically stable log(1+exp(x))
    return fmaxf(x, 0.0f) + log1pf(__expf(-fabsf(x)));
}
__device__ __forceinline__ float sigmoidf(float x) {
    return 1.0f / (1.0f + __expf(-x));
}

// ---------------------------------------------------------------------------
// Kernel 1: fold gate*scale*beta into w[1024] and compute the KL scalar.
// Tiny (1024 elems) — one workgroup, LDS tree reduction.
// ---------------------------------------------------------------------------
__global__ __launch_bounds__(256) void prep_kernel(
    const int* __restrict__ training_p,
    const float* __restrict__ eps_beta, const float* __restrict__ eps_sigma,
    const float* __restrict__ unif,
    const float* __restrict__ mu_beta, const float* __restrict__ rho_beta,
    const float* __restrict__ p_logit,
    const float* __restrict__ mu_logsigma, const float* __restrict__ rho_logsigma,
    const float* __restrict__ pi_post,
    float* __restrict__ w_out, float* __restrict__ kl_out)
{
    __shared__ float red[256];
    const int t = threadIdx.x;
    const int training = training_p[0];
    const float pi    = pi_post[0];
    const float lpi   = __logf(pi + EPS);
    const float l1mpi = __logf(1.0f - pi + EPS);

    float acc = 0.0f;
    for (int j = 0; j < D_DIM; j += 256) {
        const int d = j + t;
        const float mb = mu_beta[d];
        const float vb = softplusf(rho_beta[d]);        // var_beta
        const float ml = mu_logsigma[d];
        const float vs = softplusf(rho_logsigma[d]);    // var_sigma
        const float p  = sigmoidf(p_logit[d]);

        float w;
        if (training) {
            const float beta = mb + vb * eps_beta[d];
            const float sig  = __expf(ml + vs * eps_sigma[d]);
            const float u    = unif[d];
            // hard-concrete relaxation gate, TEMP = 0.3
            const float s = __logf(p + EPS_HC) - __logf(1.0f - p + EPS_HC)
                          + __logf(u + EPS_HC) - __logf(1.0f - u + EPS_HC);
            const float keep = sigmoidf(s * (1.0f / 0.3f));
            w = keep * sig * beta;
        } else {
            w = p * __expf(ml) * mb;
        }
        w_out[d] = w;

        // kl_beta:  (vb^2+mb^2)/(2*0.36) - log(vb+eps) + log(0.6) - 0.5
        float kl = (vb * vb + mb * mb) * (0.5f / (0.6f * 0.6f))
                 - __logf(vb + EPS) + (-0.51082562376f) - 0.5f;
        // spike entropy + expected prior (pi scalar)
        kl += p * __logf(p + EPS) + (1.0f - p) * __logf(1.0f - p + EPS);
        kl += -(p * lpi + (1.0f - p) * l1mpi);
        // kl_scale = -entropy_sigma - prior_sigma
        kl -= ml + __logf(vs) + 0.91893853320f + 0.5f;   // 0.5*log(2*pi)
        // prior_sigma term: ALPHA_G=1, BETA_G=0.5, lgamma(1)=0
        kl -= (-0.69314718056f) - 2.0f * ml
              - 0.5f * __expf(-ml + 0.5f * vs * vs);
        acc += kl;
    }
    red[t] = acc;
    __syncthreads();
    for (int s = 128; s > 0; s >>= 1) {
        if (t < s) red[t] += red[t + s];
        __syncthreads();
    }
    // exp_beta term with ALPHA_B=BETA_B=1: (0)*log(pi+eps)+(0)*log(1-pi+eps)
    // - lgamma(1) - lgamma(1) + lgamma(2) == 0 exactly.
    if (t == 0) kl_out[0] = red[0];
}

// ---------------------------------------------------------------------------
// Kernel 2: out[n] = dot(x[n,:], w) via V_WMMA_F32_16X16X4_F32.
// 8 waves/block, 16 rows/wave. B operand = w[k..k+3] broadcast to all 16
// columns -> every column of D equals the row dot product.
// A layout (32-bit 16x4): lanes 0-15 rows M=0..15 with VGPR0=K0,VGPR1=K1;
// lanes 16-31 same rows with K2,K3  ==> one global_load_b64 per lane.
// B layout mirrors it ==> one broadcast ds_load_b64 per lane from LDS.
// ---------------------------------------------------------------------------
__global__ __launch_bounds__(256) void gemv_wmma_kernel(
    const float* __restrict__ x, const float* __restrict__ w,
    float* __restrict__ out)
{
    __shared__ float wlds[D_DIM];
    const int t = threadIdx.x;
    // stage w (4 KB) into LDS, one b128 per thread
    *(v4f*)(&wlds[t * 4]) = *(const v4f*)(w + t * 4);
    __syncthreads();

    const int  lane = t & 31;
    const int  wv   = t >> 5;
    const int  hi   = lane >> 4;               // K-pair select (0 -> K0K1, 1 -> K2K3)
    const long rowBase = (long)blockIdx.x * 128 + (long)wv * 16;
    const float* ap = x + (rowBase + (lane & 15)) * (long)D_DIM + hi * 2;
    const float* bp = wlds + hi * 2;

    v8f acc = {0.f, 0.f, 0.f, 0.f, 0.f, 0.f, 0.f, 0.f};
#pragma unroll 4
    for (int k = 0; k < D_DIM; k += 4) {
        v2f a = *(const v2f*)(ap + k);         // x tile, WMMA A layout
        v2f b = *(const v2f*)(bp + k);         // w broadcast, WMMA B layout
        // D = A(16x4,f32) * B(4x16,f32) + C ; full-EXEC here (no divergence yet)
        acc = __builtin_amdgcn_wmma_f32_16x16x4_f32(
                  false, a, false, b, (short)0, acc, false, false);
    }

    // D layout: VGPR i, lanes 0-15 => row M=i; lanes 16-31 => row M=8+i.
    // All columns identical, so lane 0 holds rows 0..7, lane 16 rows 8..15.
    if ((lane & 15) == 0) {
        float* op = out + rowBase + hi * 8;    // 16B aligned (rowBase % 16 == 0)
        v4f lo4 = {acc[0], acc[1], acc[2], acc[3]};
        v4f hi4 = {acc[4], acc[5], acc[6], acc[7]};
        *(v4f*)op       = lo4;
        *(v4f*)(op + 4) = hi4;
    }
}

// ---------------------------------------------------------------------------
extern "C" void kernel_launch(void* const* d_in, const int* in_sizes, int n_in,
                              void* d_out, int out_size, void* d_ws, size_t ws_size,
                              hipStream_t stream) {
    const float* x            = (const float*)d_in[0];
    const int*   training     = (const int*)  d_in[1];
    const float* eps_beta     = (const float*)d_in[2];
    const float* eps_sigma    = (const float*)d_in[3];
    const float* unif         = (const float*)d_in[4];
    const float* mu_beta      = (const float*)d_in[5];
    const float* rho_beta     = (const float*)d_in[6];
    const float* p_logit      = (const float*)d_in[7];
    const float* mu_logsigma  = (const float*)d_in[8];
    const float* rho_logsigma = (const float*)d_in[9];
    const float* pi_post      = (const float*)d_in[10];

    float* out = (float*)d_out;
    float* w   = (float*)d_ws;                 // 1024 floats of scratch
    const int N = in_sizes[0] / D_DIM;         // 262144

    prep_kernel<<<1, 256, 0, stream>>>(training, eps_beta, eps_sigma, unif,
                                       mu_beta, rho_beta, p_logit,
                                       mu_logsigma, rho_logsigma, pi_post,
                                       w, out + (size_t)N);

    gemv_wmma_kernel<<<N / 128, 256, 0, stream>>>(x, w, out);
}